// CurveGraphic2d_34617436405811
// MI455X (gfx1250) — compile-verified
//
#include <hip/hip_runtime.h>
#include <math.h>

typedef __attribute__((ext_vector_type(2))) float v2f;
typedef __attribute__((ext_vector_type(8))) float v8f;

#define CANVAS_H 256
#define CANVAS_W 256
#define NSAMP    32      // bezier samples per curve
#define NCURVE   32      // batch
#define BWIDTH   4.0f
#define AAF      0.35f
#define EPS      1e-6f

// --- raw hardware transcendentals (v_sqrt_f32 / v_log_f32 / v_exp_f32) -----
// Bodies must also parse in the host pass, where amdgcn builtins don't exist.
static __device__ __forceinline__ float hw_sqrt(float x) {
#if defined(__HIP_DEVICE_COMPILE__) && __has_builtin(__builtin_amdgcn_sqrtf)
  return __builtin_amdgcn_sqrtf(x);
#elif defined(__HIP_DEVICE_COMPILE__)
  return __fsqrt_rn(x);
#else
  return sqrtf(x);
#endif
}
static __device__ __forceinline__ float hw_log2(float x) {
#if defined(__HIP_DEVICE_COMPILE__) && __has_builtin(__builtin_amdgcn_logf)
  return __builtin_amdgcn_logf(x);
#elif defined(__HIP_DEVICE_COMPILE__)
  return __log2f(x);
#else
  return log2f(x);
#endif
}
static __device__ __forceinline__ float hw_exp2(float x) {
#if defined(__HIP_DEVICE_COMPILE__) && __has_builtin(__builtin_amdgcn_exp2f)
  return __builtin_amdgcn_exp2f(x);
#elif defined(__HIP_DEVICE_COMPILE__)
  return __expf(x * 0.6931471805599453f);   // native exp path
#else
  return exp2f(x);
#endif
}

// ---------------------------------------------------------------------------
// Preamble: evaluate the cubic Bernstein basis at t = s/31 and emit the
// WMMA A-operand image directly in the 16x4 f32 lane layout:
//   ws[b*64 + tile*32 + lane]  (float2)
//     lane 0-15  -> { s0, s1 }        (K=0, K=1 of sample m = tile*16+lane)
//     lane 16-31 -> { 1.0, ||s||^2 }  (K=2, K=3 of sample m = tile*16+lane-16)
// 2048 entries, 16 KB. Main kernel then does a bare coalesced b64 load.
// ---------------------------------------------------------------------------
__global__ __launch_bounds__(256) void curve_samples_kernel(
    const float* __restrict__ in, float2* __restrict__ ws) {
  int id = blockIdx.x * blockDim.x + threadIdx.x;   // 0..2047
  int b    = id >> 6;          // curve
  int tile = (id >> 5) & 1;    // sample tile (s 0-15 / 16-31)
  int lane = id & 31;
  int s = tile * 16 + (lane & 15);

  float t = (float)s * (1.0f / 31.0f);
  float u = 1.0f - t;
  float b0 = u * u * u;
  float b1 = 3.0f * t * u * u;
  float b2 = 3.0f * t * t * u;
  float b3 = t * t * t;
  const float* kp = in + b * 8;                      // [K=4][2], normalized
  float s0 = 256.0f * (b0 * kp[0] + b1 * kp[2] + b2 * kp[4] + b3 * kp[6]);
  float s1 = 256.0f * (b0 * kp[1] + b1 * kp[3] + b2 * kp[5] + b3 * kp[7]);

  float2 v;
  if (lane < 16) { v.x = s0;   v.y = s1; }
  else           { v.x = 1.0f; v.y = s0 * s0 + s1 * s1; }
  ws[id] = v;
}

// ---------------------------------------------------------------------------
// Main kernel: one wave32 owns a 16-pixel horizontal tile of the canvas.
// d2 = A(16 samples x 4) * B(4 x 16 pixels) via V_WMMA_F32_16X16X4_F32:
//   A[m] = [ s0, s1, 1, ||s||^2 ]     (M = sample)
//   B[n] = [ -2*y, -2*x, ||p||^2, 1 ] (N = pixel)
// Two curves per iteration (4 WMMAs); per-curve min over 16 accumulator
// values (tree) + xor-16 shuffle; then lanes 0-15 carry curve b, lanes 16-31
// carry curve b+1 and the whole wave stores 32 pixels — no exec masking,
// EXEC stays all-ones for the entire kernel (required for WMMA anyway).
// ---------------------------------------------------------------------------
__global__ __launch_bounds__(256) void curve_render_kernel(
    const float2* __restrict__ sampA, float* __restrict__ out) {
  const int lane  = threadIdx.x & 31;
  const int wave  = blockIdx.x * (blockDim.x >> 5) + (threadIdx.x >> 5); // 0..4095
  const int y     = wave >> 4;              // canvas row
  const int x0    = (wave & 15) << 4;       // 16-pixel chunk start
  const int n     = lane & 15;              // pixel column within tile
  const bool hi   = lane >= 16;

  // B operand (pixels), built once per wave (4x16 f32 lane layout).
  const float px = (float)(x0 + n);
  const float py = (float)y;
  const float p2 = px * px + py * py;
  v2f Bv;
  Bv.x = hi ? p2   : -2.0f * py;   // K=0: -2*p_y   | K=2: ||p||^2
  Bv.y = hi ? 1.0f : -2.0f * px;   // K=1: -2*p_x   | K=3: 1

  const unsigned outBase = (unsigned)y * CANVAS_W + x0 + n + (hi ? CANVAS_H * CANVAS_W : 0);

  for (int b = 0; b < NCURVE; b += 2) {
    // A operands: pre-formatted, one b64 load per (curve, tile).
    float2 a00 = sampA[(b    ) * 64 +      lane];
    float2 a01 = sampA[(b    ) * 64 + 32 + lane];
    float2 a10 = sampA[(b + 1) * 64 +      lane];
    float2 a11 = sampA[(b + 1) * 64 + 32 + lane];

    v2f A00 = {a00.x, a00.y}, A01 = {a01.x, a01.y};
    v2f A10 = {a10.x, a10.y}, A11 = {a11.x, a11.y};

    v8f c00 = {}, c01 = {}, c10 = {}, c11 = {};
    c00 = __builtin_amdgcn_wmma_f32_16x16x4_f32(false, A00, false, Bv, (short)0, c00, false, false);
    c01 = __builtin_amdgcn_wmma_f32_16x16x4_f32(false, A01, false, Bv, (short)0, c01, false, false);
    c10 = __builtin_amdgcn_wmma_f32_16x16x4_f32(false, A10, false, Bv, (short)0, c10, false, false);
    c11 = __builtin_amdgcn_wmma_f32_16x16x4_f32(false, A11, false, Bv, (short)0, c11, false, false);

    // Tree min over the 16 sample rows this lane sees, per curve.
    float m0[8], m1[8];
#pragma unroll
    for (int j = 0; j < 8; ++j) m0[j] = fminf(c00[j], c01[j]);
#pragma unroll
    for (int j = 0; j < 8; ++j) m1[j] = fminf(c10[j], c11[j]);
    float d0 = fminf(fminf(fminf(m0[0], m0[4]), fminf(m0[1], m0[5])),
                     fminf(fminf(m0[2], m0[6]), fminf(m0[3], m0[7])));
    float d1 = fminf(fminf(fminf(m1[0], m1[4]), fminf(m1[1], m1[5])),
                     fminf(fminf(m1[2], m1[6]), fminf(m1[3], m1[7])));

    // lanes l and l+16 hold the same pixel N with disjoint sample halves
    d0 = fminf(d0, __shfl_xor(d0, 16, 32));
    d1 = fminf(d1, __shfl_xor(d1, 16, 32));

    // lanes 0-15 -> curve b, lanes 16-31 -> curve b+1; full-wave tail+store
    float dmin = hi ? d1 : d0;
    float d   = hw_sqrt(fmaxf(dmin, 0.0f));
    float arg = d * (1.0f / BWIDTH) + EPS;           // > 0 always
    float v   = 1.0f - hw_exp2(AAF * hw_log2(arg));
    out[(unsigned)b * (CANVAS_H * CANVAS_W) + outBase] = v;
  }
}

// ---------------------------------------------------------------------------
extern "C" void kernel_launch(void* const* d_in, const int* in_sizes, int n_in,
                              void* d_out, int out_size, void* d_ws, size_t ws_size,
                              hipStream_t stream) {
  const float* inputs = (const float*)d_in[0];     // [32,4,2] float32
  float2* samples = (float2*)d_ws;                 // 2048 * 8B = 16 KB
  float* out = (float*)d_out;                      // [32,256,256] float32

  curve_samples_kernel<<<8, 256, 0, stream>>>(inputs, samples);

  // 4096 pixel tiles, 8 waves per 256-thread block -> 512 blocks
  curve_render_kernel<<<512, 256, 0, stream>>>(samples, out);
}